// NeuronMoEDecoderLayer_40450001994264
// MI455X (gfx1250) — compile-verified
//
#include <hip/hip_runtime.h>
#include <hip/hip_bf16.h>

// MoE decoder layer for MI455X (gfx1250, wave32).
// H=1024, I=512, E=16, K=4, T=B*S=2048. All fp32.
// fp32 WMMA (V_WMMA_F32_16X16X4_F32): full precision; workload is L2/HBM
// bound (weights = 107MB fit in 192MB L2), so fp32 matrix rate is ample.

typedef __attribute__((ext_vector_type(2))) float v2f;
typedef __attribute__((ext_vector_type(8))) float v8f;

#define HDIM 1024
#define IDIM 512
#define NEXP 16
#define TOPK 4
#define NTOK 2048   // B*S
#define NTILE (NTOK / 16)

__device__ __forceinline__ float fast_sigmoid(float z) {
    // 1/(1+exp(-z)) with v_rcp_f32 instead of the IEEE divide chain
    return __builtin_amdgcn_rcpf(1.f + __expf(-z));
}

// ---------------------------------------------------------------------------
// Kernel 1: router. One block of 256 threads per 16-token tile.
// thread = (token = tid/16, expert = tid%16). Computes sigmoid(x @ router_w),
// exact top-4 with lowest-index tie-break, writes dense route[T,E] and a
// per-tile bitmask of experts selected by any token in the tile.
// ---------------------------------------------------------------------------
__global__ __launch_bounds__(256)
void moe_router_kernel(const float* __restrict__ x,
                       const float* __restrict__ rw,
                       float* __restrict__ route,
                       unsigned* __restrict__ tile_mask)
{
    __shared__ float aff[16][16];
    __shared__ unsigned selm[16];

    const int tid = threadIdx.x;
    const int tok = tid >> 4;     // 0..15
    const int e   = tid & 15;     // 0..15
    const int t   = blockIdx.x * 16 + tok;

    const float* xr = x + (size_t)t * HDIM;
    float acc = 0.f;
#pragma unroll 8
    for (int h = 0; h < HDIM; ++h)
        acc = fmaf(xr[h], rw[h * NEXP + e], acc);

    aff[tok][e] = fast_sigmoid(acc);
    __syncthreads();

    if (tid < 16) {
        // top-4 of 16 affinities for token `tid` (strict > keeps lowest index)
        unsigned sel = 0;
        for (int it = 0; it < TOPK; ++it) {
            float best = -1.f; int bi = 0;
            for (int j = 0; j < NEXP; ++j) {
                if (!((sel >> j) & 1u) && aff[tid][j] > best) {
                    best = aff[tid][j]; bi = j;
                }
            }
            sel |= 1u << bi;
        }
        float* rr = route + (size_t)(blockIdx.x * 16 + tid) * NEXP;
        for (int j = 0; j < NEXP; ++j)
            rr[j] = ((sel >> j) & 1u) ? aff[tid][j] : 0.f;
        selm[tid] = sel;
    }
    __syncthreads();

    if (tid == 0) {
        unsigned m = 0;
        for (int j = 0; j < 16; ++j) m |= selm[j];
        tile_mask[blockIdx.x] = m;
    }
}

// ---------------------------------------------------------------------------
// Kernel 2: fused expert pipeline. One block (8 waves / 256 threads) per
// 16-token tile. Loops over {shared expert, active routed experts}; per
// expert: phase 1 = gate/up GEMMs (fp32 WMMA, A scaled by route), SiLU*up
// into LDS; phase 2 = down-projection accumulated into persistent v8f regs.
// ---------------------------------------------------------------------------
__global__ __launch_bounds__(256)
void moe_expert_kernel(const float* __restrict__ x,
                       const float* __restrict__ gate_w,   // [E][H][I]
                       const float* __restrict__ up_w,     // [E][H][I]
                       const float* __restrict__ down_w,   // [E][I][H]
                       const float* __restrict__ sgw,      // [H][I]
                       const float* __restrict__ suw,      // [H][I]
                       const float* __restrict__ sdw,      // [I][H]
                       const float* __restrict__ route,    // [T][E]
                       const unsigned* __restrict__ tile_mask,
                       float* __restrict__ out)            // [T][H]
{
    // +4 float padding -> row r shifts banks by 4r: 16-row column reads are
    // conflict-free across all 32 lanes.
    __shared__ float xs[16][HDIM + 4];   // x tile        (~65.8 KB)
    __shared__ float hs[16][IDIM + 4];   // silu(g)*u     (~33.0 KB)

    const int tid   = threadIdx.x;
    const int lane  = tid & 31;
    const int wave  = tid >> 5;          // 0..7
    const int half  = lane >> 4;         // 0 or 1
    const int mrow  = lane & 15;         // 0..15 (M row / N col within tile)
    const int tile0 = blockIdx.x * 16;

    // ---- stage x tile into LDS (coalesced float4 loads) ----
    const float4* xv = reinterpret_cast<const float4*>(x + (size_t)tile0 * HDIM);
#pragma unroll
    for (int i = 0; i < 16; ++i) {
        int idx = i * 256 + tid;         // float4 index, 16 rows * 256/row
        int r = idx >> 8, c4 = idx & 255;
        float4 v = xv[idx];
        float* d = &xs[r][c4 * 4];
        d[0] = v.x; d[1] = v.y; d[2] = v.z; d[3] = v.w;
    }
    __syncthreads();

    const unsigned mask = tile_mask[blockIdx.x];

    v8f accy[8];
#pragma unroll
    for (int j = 0; j < 8; ++j)
        accy[j] = (v8f){0.f, 0.f, 0.f, 0.f, 0.f, 0.f, 0.f, 0.f};

    // eidx 0 = always-on shared expert; eidx 1..16 = routed experts (skippable)
    for (int eidx = 0; eidx <= NEXP; ++eidx) {
        const float *gw, *uw, *dw;
        float rv;
        if (eidx == 0) {
            gw = sgw; uw = suw; dw = sdw; rv = 1.f;
        } else {
            const int e = eidx - 1;
            if (!((mask >> e) & 1u)) continue;     // uniform scalar branch
            gw = gate_w + (size_t)e * HDIM * IDIM;
            uw = up_w   + (size_t)e * HDIM * IDIM;
            dw = down_w + (size_t)e * IDIM * HDIM;
            rv = route[(size_t)(tile0 + mrow) * NEXP + e];  // per-row scale
        }

        // ---- phase 1: G/U = (rv*x) @ W, h = silu(G)*U -> hs ----
        // wave handles I-columns [wave*64, wave*64+64)
#pragma unroll
        for (int j = 0; j < 4; ++j) {
            const int nb = wave * 64 + j * 16;
            v8f ag = (v8f){0.f,0.f,0.f,0.f,0.f,0.f,0.f,0.f};
            v8f au = (v8f){0.f,0.f,0.f,0.f,0.f,0.f,0.f,0.f};
            const float* gp = gw + (size_t)(2 * half) * IDIM + nb + mrow;
            const float* up = uw + (size_t)(2 * half) * IDIM + nb + mrow;
            const float* ap = &xs[mrow][2 * half];
#pragma unroll 4
            for (int k = 0; k < HDIM; k += 4) {
                v2f a = *(const v2f*)(ap + k);     // ds_load_b64, bank-safe
                a.x *= rv; a.y *= rv;
                v2f bg; bg.x = gp[0]; bg.y = gp[IDIM];
                v2f bu; bu.x = up[0]; bu.y = up[IDIM];
                ag = __builtin_amdgcn_wmma_f32_16x16x4_f32(
                         false, a, false, bg, (short)0, ag, false, false);
                au = __builtin_amdgcn_wmma_f32_16x16x4_f32(
                         false, a, false, bu, (short)0, au, false, false);
                gp += 4 * IDIM; up += 4 * IDIM;
            }
#pragma unroll
            for (int r = 0; r < 8; ++r) {
                float g = ag[r], u = au[r];
                float hv = g * fast_sigmoid(g) * u;        // silu(g)*u
                hs[r + 8 * half][nb + mrow] = hv;
            }
        }
        __syncthreads();

        // ---- phase 2: accy += h @ down_w ----
        // wave handles H-columns [wave*128, wave*128+128)
#pragma unroll
        for (int j = 0; j < 8; ++j) {
            const int nb = wave * 128 + j * 16;
            const float* dp = dw + (size_t)(2 * half) * HDIM + nb + mrow;
            const float* ap = &hs[mrow][2 * half];
            v8f acc = accy[j];
#pragma unroll 4
            for (int k = 0; k < IDIM; k += 4) {
                v2f a = *(const v2f*)(ap + k);
                v2f b; b.x = dp[0]; b.y = dp[HDIM];
                acc = __builtin_amdgcn_wmma_f32_16x16x4_f32(
                          false, a, false, b, (short)0, acc, false, false);
                dp += 4 * HDIM;
            }
            accy[j] = acc;
        }
        __syncthreads();   // hs reused by next expert
    }

    // ---- write output tile (each (t,h) owned by exactly one lane) ----
#pragma unroll
    for (int j = 0; j < 8; ++j) {
        const int nb = wave * 128 + j * 16;
#pragma unroll
        for (int r = 0; r < 8; ++r)
            out[(size_t)(tile0 + r + 8 * half) * HDIM + nb + mrow] = accy[j][r];
    }
}

// ---------------------------------------------------------------------------
extern "C" void kernel_launch(void* const* d_in, const int* in_sizes, int n_in,
                              void* d_out, int out_size, void* d_ws, size_t ws_size,
                              hipStream_t stream) {
    (void)in_sizes; (void)n_in; (void)out_size; (void)ws_size;

    const float* x   = (const float*)d_in[0];   // [B,S,H]
    const float* rw  = (const float*)d_in[1];   // [H,E]
    const float* gw  = (const float*)d_in[2];   // [E,H,I]
    const float* uw  = (const float*)d_in[3];   // [E,H,I]
    const float* dw  = (const float*)d_in[4];   // [E,I,H]
    const float* sgw = (const float*)d_in[5];   // [H,I]
    const float* suw = (const float*)d_in[6];   // [H,I]
    const float* sdw = (const float*)d_in[7];   // [I,H]
    float* out = (float*)d_out;                 // [B,S,H]

    float* route        = (float*)d_ws;                                   // T*E f32
    unsigned* tile_mask = (unsigned*)((char*)d_ws + (size_t)NTOK * NEXP * sizeof(float));

    moe_router_kernel<<<NTILE, 256, 0, stream>>>(x, rw, route, tile_mask);
    moe_expert_kernel<<<NTILE, 256, 0, stream>>>(x, gw, uw, dw, sgw, suw, sdw,
                                                 route, tile_mask, out);
}